// CTCLayer_16466904612902
// MI455X (gfx1250) — compile-verified
//
#include <hip/hip_runtime.h>
#include <hip/hip_bf16.h>
#include <stdint.h>

// ---------------------------------------------------------------------------
// CTC forward loss, CDNA5 (gfx1250).
//   B blocks x 32 lanes (one wave32 per batch element).
//   Lane l owns extended states 4l..4l+3; lane 31 also owns state 128.
//   y_pred tiles streamed HBM -> LDS by the Tensor Data Mover (TENSORcnt),
//   double-buffered; cross-lane alpha dependency via ds_bpermute (no barriers).
// ---------------------------------------------------------------------------

#define CTC_C      128          // classes (blank = 127)
#define CTC_L      64           // max label length  -> S = 129 states
#define TILE_ROWS  32           // time rows per TDM tile (32*128*4 = 16 KB)
#define NEGF       (-1e30f)
#define EPSF       (1e-7f)

#if __has_builtin(__builtin_amdgcn_tensor_load_to_lds)
#define CTC_HAS_TDM 1
#endif

typedef unsigned int u32x4 __attribute__((ext_vector_type(4)));
typedef int          i32x8 __attribute__((ext_vector_type(8)));
typedef int          i32x4 __attribute__((ext_vector_type(4)));

__device__ __forceinline__ float ladd(float a, float b) {
    // logaddexp, robust for NEGF sentinels (exp underflows to 0)
    float mx = fmaxf(a, b);
    float d  = fminf(a, b) - mx;
    return mx + __logf(1.0f + __expf(d));
}

__device__ __forceinline__ float shup1(float v, float fill, int lane) {
    int r = __builtin_amdgcn_ds_bpermute((lane - 1) << 2, __float_as_int(v));
    return (lane == 0) ? fill : __int_as_float(r);
}
__device__ __forceinline__ int shup1_i(int v, int lane) {
    return __builtin_amdgcn_ds_bpermute((lane - 1) << 2, v);
}

// Issue one TDM load: 'rows' consecutive rows of 128 f32 -> LDS buffer.
__device__ __forceinline__ void tdm_issue(float* ldsbuf, const float* gsrc, int rows) {
#ifdef CTC_HAS_TDM
    uint64_t ga = (uint64_t)(uintptr_t)gsrc;
    uint32_t lo = (uint32_t)(uintptr_t)ldsbuf;        // low 32 bits = LDS byte offset
    u32x4 g0;
    g0.x = 1u;                                        // count=1, user descriptor
    g0.y = lo;                                        // lds_addr (bytes)
    g0.z = (uint32_t)ga;                              // global_addr[31:0]
    g0.w = ((uint32_t)(ga >> 32) & 0x01FFFFFFu)       // global_addr[56:32]
         | (2u << 30);                                // type = 2 ("image")
    i32x8 g1;
    g1[0] = 0x00020000;                               // data_size = 4 bytes
    g1[1] = (int)(128u << 16);                        // tensor_dim0 = 128 (lo16)
    g1[2] = (int)(512u << 16);                        // dim0 hi=0 | tensor_dim1 = 512 (lo16)
    g1[3] = (int)(128u << 16);                        // dim1 hi=0 | tile_dim0 = 128
    g1[4] = rows & 0xFFFF;                            // tile_dim1 = rows | tile_dim2 = 0
    g1[5] = 128;                                      // tensor_dim0_stride[31:0] = 128
    g1[6] = 0;                                        // stride hi | dim1_stride lo (2D: unused)
    g1[7] = 0;
    i32x4 z4 = 0;                                     // group2 (unused, 2D tensor)
    i32x8 z8 = 0;                                     // group3/extra (unused)
    __builtin_amdgcn_tensor_load_to_lds(g0, g1, z4, z4, z8, 0);
#else
    // Fallback: cooperative wave copy (keeps kernel functional if builtin absent)
    int lane = threadIdx.x & 31;
    for (int i = lane; i < rows * CTC_C; i += 32) ldsbuf[i] = gsrc[i];
#endif
}

__device__ __forceinline__ void tdm_wait_le1() {
#ifdef CTC_HAS_TDM
#if __has_builtin(__builtin_amdgcn_s_wait_tensorcnt)
    __builtin_amdgcn_s_wait_tensorcnt(1);
#else
    asm volatile("s_wait_tensorcnt 0x1" ::: "memory");
#endif
#endif
}
__device__ __forceinline__ void tdm_wait_le0() {
#ifdef CTC_HAS_TDM
#if __has_builtin(__builtin_amdgcn_s_wait_tensorcnt)
    __builtin_amdgcn_s_wait_tensorcnt(0);
#else
    asm volatile("s_wait_tensorcnt 0x0" ::: "memory");
#endif
#endif
}

__global__ void __launch_bounds__(32)
ctc_loss_kernel(const float* __restrict__ y_pred,   // [B, T, 128]
                const int*   __restrict__ labels,   // [B, 64]
                const int*   __restrict__ input_length,
                const int*   __restrict__ label_length,
                float*       __restrict__ out,      // [B]
                int T)
{
    __shared__ float tiles[2][TILE_ROWS * CTC_C];   // 32 KB double buffer
    __shared__ float afin[2 * CTC_L + 4];           // final alpha spill

    const int b    = blockIdx.x;
    const int lane = threadIdx.x & 31;

    const float* yp = y_pred + (size_t)b * T * CTC_C;
    const int*   lb = labels + (size_t)b * CTC_L;

    // Each lane owns labels 2l and 2l+1 (states 4l+1 and 4l+3).
    int lab0 = lb[2 * lane]     & (CTC_C - 1);
    int lab1 = lb[2 * lane + 1] & (CTC_C - 1);
    int labp = shup1_i(lab1, lane);                 // labels[2l-1] from lane l-1
    // skip(s) = ext[s] != blank && ext[s] != ext[s-2]
    bool skip1 = (lab0 != CTC_C - 1) && ((lane == 0) || (lab0 != labp));
    bool skip3 = (lab1 != CTC_C - 1) && (lab1 != lab0);

    int Tb = input_length[b];
    Tb = Tb < 1 ? 1 : (Tb > T ? T : Tb);            // alpha frozen past input_length
    int ll = label_length[b];
    ll = ll < 1 ? 1 : (ll > CTC_L ? CTC_L : ll);
    const int sl = 2 * ll;

    const int ntiles = (Tb + TILE_ROWS - 1) / TILE_ROWS;

    // Prefetch tile 0.
    {
        int rows0 = Tb < TILE_ROWS ? Tb : TILE_ROWS;
        tdm_issue(tiles[0], yp, rows0);
    }

    float a0 = NEGF, a1 = NEGF, a2 = NEGF, a3 = NEGF, a4 = NEGF;

    for (int k = 0; k < ntiles; ++k) {
        const int tstart = k * TILE_ROWS;
        int rows = Tb - tstart; if (rows > TILE_ROWS) rows = TILE_ROWS;

        if (k + 1 < ntiles) {
            const int ns = tstart + TILE_ROWS;
            int nrows = Tb - ns; if (nrows > TILE_ROWS) nrows = TILE_ROWS;
            tdm_issue(tiles[(k + 1) & 1], yp + (size_t)ns * CTC_C, nrows);
            tdm_wait_le1();                         // in-order: tile k complete
        } else {
            tdm_wait_le0();
        }
        __syncthreads();                            // LDS visibility / compiler fence

        const float* bp = tiles[k & 1];
        int r = 0;
        if (k == 0) {                               // t = 0 initialization
            float lpb = __logf(bp[CTC_C - 1] + EPSF);
            float lp1 = __logf(bp[lab0]      + EPSF);
            a0 = (lane == 0) ? lpb : NEGF;          // alpha[0] = lp(blank)
            a1 = (lane == 0) ? lp1 : NEGF;          // alpha[1] = lp(label0)
            r = 1;
        }
        for (; r < rows; ++r) {
            const float* row = bp + r * CTC_C;
            float lpb = __logf(row[CTC_C - 1] + EPSF);  // broadcast: blank prob
            float lp1 = __logf(row[lab0]      + EPSF);
            float lp3 = __logf(row[lab1]      + EPSF);
            float p3  = shup1(a3, NEGF, lane);          // alpha[4l-1] from lane l-1

            float n0 = ladd(a0, p3) + lpb;                              // s = 4l   (blank)
            float t1 = ladd(a1, a0);
            float n1 = (skip1 ? ladd(t1, p3) : t1) + lp1;               // s = 4l+1
            float n2 = ladd(a2, a1) + lpb;                              // s = 4l+2 (blank)
            float t3 = ladd(a3, a2);
            float n3 = (skip3 ? ladd(t3, a1) : t3) + lp3;               // s = 4l+3
            float n4 = ladd(a4, a3) + lpb;                              // s = 128 (lane 31)

            a0 = n0; a1 = n1; a2 = n2; a3 = n3; a4 = n4;
        }
    }

    // Gather final alpha and emit loss.
    afin[4 * lane + 0] = a0;
    afin[4 * lane + 1] = a1;
    afin[4 * lane + 2] = a2;
    afin[4 * lane + 3] = a3;
    if (lane == 31) afin[128] = a4;
    __syncthreads();
    if (lane == 0) {
        out[b] = -ladd(afin[sl], afin[sl - 1]);
    }
}

extern "C" void kernel_launch(void* const* d_in, const int* in_sizes, int n_in,
                              void* d_out, int out_size, void* d_ws, size_t ws_size,
                              hipStream_t stream) {
    (void)n_in; (void)d_ws; (void)ws_size;
    const float* y_pred       = (const float*)d_in[0];   // [B, T, C] f32
    const int*   labels       = (const int*)d_in[1];     // [B, L] i32
    const int*   input_length = (const int*)d_in[2];     // [B] i32
    const int*   label_length = (const int*)d_in[3];     // [B] i32
    float*       out          = (float*)d_out;           // [B, 1] f32

    const int B = in_sizes[2];
    const int T = in_sizes[0] / (B * CTC_C);

    ctc_loss_kernel<<<dim3(B), dim3(32), 0, stream>>>(
        y_pred, labels, input_length, label_length, out, T);
}